// LearnableTD_79800492359911
// MI455X (gfx1250) — compile-verified
//
#include <hip/hip_runtime.h>
#include <stdint.h>
#include <math.h>

// ---------------------------------------------------------------------------
// CDNA5 async global->LDS staging (ASYNCcnt path). Inline asm is used since
// the clang builtins for async-LDS differ across toolchains; the ISA mnemonics
// (CDNA5 ISA opcodes 96/98, SOPP 74) are stable.
// GVS addressing: mem = SADDR(64b sgpr pair) + VADDR(32b per-lane byte off).
// VDST = per-lane LDS byte address.
// ---------------------------------------------------------------------------
#define ASYNC_LDS_B128(ldsaddr, voff, sbase)                                   \
  asm volatile("global_load_async_to_lds_b128 %0, %1, %2"                      \
               :: "v"(ldsaddr), "v"(voff), "s"(sbase) : "memory")
#define ASYNC_LDS_B32(ldsaddr, voff, sbase)                                    \
  asm volatile("global_load_async_to_lds_b32 %0, %1, %2"                       \
               :: "v"(ldsaddr), "v"(voff), "s"(sbase) : "memory")
// ASYNCcnt decrements in issue order, so waiting <=6 after issuing the next
// 6-op group guarantees the previous group's LDS writes have landed.
#define WAIT_ASYNC_LE6() asm volatile("s_wait_asynccnt 0x6" ::: "memory")
#define WAIT_ASYNC_0()   asm volatile("s_wait_asynccnt 0x0" ::: "memory")

#ifndef TD_MAX_T
#define TD_MAX_T 1024
#endif

// Native vector type (not HIP_vector_type) so __builtin_nontemporal_store
// accepts it and it lowers to a single global_store_b128 th:NT.
typedef float v4f __attribute__((ext_vector_type(4)));

// One wave32 per batch row; 128 timesteps per chunk (4 per lane).
// Backward affine-map suffix scan:
//   lr_t = a_t*lr_{t+1} + u_t ,  sr_t = a_t*sr_{t+1} + r_t   (shared a_t)
// Lane-local compose (4 elems) -> wave suffix scan (5 shuffle steps) ->
// per-lane apply. Carry chained across chunks via lane-0 broadcast.
// lam[t] = max(tanh(raw_lambd[s0+t]),0) is row-invariant: computed once per
// block into LDS instead of 8x (once per wave) per chunk.
__global__ __launch_bounds__(256) void td_scan_kernel(
    const float* __restrict__ values,   // (B, T+1)
    const float* __restrict__ rewards,  // (B, T)
    const float* __restrict__ dones,    // (B, T)
    const float* __restrict__ raw_gamma,// (1,)
    const float* __restrict__ raw_lambd,// (T_full,)
    const int*   __restrict__ start_idx,// (1,)
    float* __restrict__ out,            // lambda_returns (B,T+1) ++ sum_rewards (B,T)
    int B, int T)
{
  __shared__ __align__(16) float s_r[8][2][128];
  __shared__ __align__(16) float s_d[8][2][128];
  __shared__ __align__(16) float s_v[8][2][128];
  __shared__ __align__(16) float s_lam[TD_MAX_T];

  const int lane = threadIdx.x & 31;
  const int w    = threadIdx.x >> 5;
  const int row  = blockIdx.x * 8 + w;

  const int s0 = start_idx[0];

  // Block-shared lam table (row-invariant): one tanh per t per block.
  for (int t = threadIdx.x; t < T; t += 256)
    s_lam[t] = fmaxf(tanhf(raw_lambd[s0 + t]), 0.0f);
  __syncthreads();                 // all waves participate before any exit

  if (row >= B) return;

  const float gamma = fmaxf(tanhf(raw_gamma[0]), 0.0f);

  float* out_lr = out;
  float* out_sr = out + (size_t)B * (size_t)(T + 1);

  // LDS byte offsets for the async writes (low 32 bits of the flat shared
  // address are the LDS offset per the gfx1250 aperture rules).
  uint32_t lds_r[2], lds_d[2], lds_v[2];
  #pragma unroll
  for (int s = 0; s < 2; ++s) {
    lds_r[s] = (uint32_t)(uintptr_t)&s_r[w][s][0];
    lds_d[s] = (uint32_t)(uintptr_t)&s_d[w][s][0];
    lds_v[s] = (uint32_t)(uintptr_t)&s_v[w][s][0];
  }

  const uint32_t row_rt = (uint32_t)row * (uint32_t)T;        // (B,T) arrays
  const uint32_t row_vt = (uint32_t)row * (uint32_t)(T + 1);  // values

  float lr_c = values[(size_t)row_vt + T];   // carry init: lr = V_T, sr = 0
  float sr_c = 0.0f;
  if (lane == 0)
    __builtin_nontemporal_store(lr_c, &out_lr[(size_t)row_vt + T]);

  const int CH = T >> 7;  // T must be a multiple of 128 on this path

  auto issue = [&](int c, int stg) {
    const int base = c << 7;
    // rewards/dones chunk: one b128 per lane (16B-aligned since T%128==0)
    const uint32_t g_rd = (row_rt + (uint32_t)base) * 4u + (uint32_t)lane * 16u;
    ASYNC_LDS_B128(lds_r[stg] + (uint32_t)lane * 16u, g_rd, rewards);
    ASYNC_LDS_B128(lds_d[stg] + (uint32_t)lane * 16u, g_rd, dones);
    // v_next chunk = values[base+1 .. base+128]: dword-aligned only, so 4x b32
    const uint32_t g_v = (row_vt + (uint32_t)base + 1u) * 4u + (uint32_t)lane * 4u;
    #pragma unroll
    for (int j = 0; j < 4; ++j)
      ASYNC_LDS_B32(lds_v[stg] + (uint32_t)(j * 32 + lane) * 4u,
                    g_v + (uint32_t)j * 128u, values);
  };

  int stg = 0;
  issue(CH - 1, stg);  // scan runs t = T-1 .. 0, so highest chunk first

  for (int c = CH - 1; c >= 0; --c) {
    if (c > 0) { issue(c - 1, stg ^ 1); WAIT_ASYNC_LE6(); }
    else       { WAIT_ASYNC_0(); }

    const int base = c << 7;

    const v4f r4 = ((const v4f*)&s_r[w][stg][0])[lane];
    const v4f d4 = ((const v4f*)&s_d[w][stg][0])[lane];
    const v4f v4 = ((const v4f*)&s_v[w][stg][0])[lane];
    const v4f l4 = ((const v4f*)&s_lam[base])[lane];   // aligned b128
    const float rr[4] = {r4.x, r4.y, r4.z, r4.w};
    const float dd[4] = {d4.x, d4.y, d4.z, d4.w};
    const float vv[4] = {v4.x, v4.y, v4.z, v4.w};
    const float ll[4] = {l4.x, l4.y, l4.z, l4.w};

    float af[4], uf[4];
    #pragma unroll
    for (int k = 0; k < 4; ++k) {
      const float cc = gamma * (1.0f - dd[k]);
      af[k] = cc * ll[k];                                  // shared multiplier
      uf[k] = fmaf(cc * (1.0f - ll[k]), vv[k], rr[k]);     // lr additive term
    }

    // Lane-local composition F = f_{4i} o f_{4i+1} o f_{4i+2} o f_{4i+3}
    float A = af[3], U = uf[3], R = rr[3];
    #pragma unroll
    for (int k = 2; k >= 0; --k) {
      U = fmaf(af[k], U, uf[k]);
      R = fmaf(af[k], R, rr[k]);
      A *= af[k];
    }

    // Wave32 inclusive suffix scan of affine maps (compose with higher lanes)
    #pragma unroll
    for (int off = 1; off < 32; off <<= 1) {
      const float Ao = __shfl_down(A, off, 32);
      const float Uo = __shfl_down(U, off, 32);
      const float Ro = __shfl_down(R, off, 32);
      if (lane + off < 32) {
        U = fmaf(A, Uo, U);
        R = fmaf(A, Ro, R);
        A *= Ao;
      }
    }

    // Exclusive map for this lane = inclusive map of lane+1 (identity @31)
    float Ae = __shfl_down(A, 1, 32);
    float Ue = __shfl_down(U, 1, 32);
    float Re = __shfl_down(R, 1, 32);
    if (lane == 31) { Ae = 1.0f; Ue = 0.0f; Re = 0.0f; }

    float lr_in = fmaf(Ae, lr_c, Ue);
    float sr_in = fmaf(Ae, sr_c, Re);

    float srv[4];
    const size_t ob = (size_t)row_vt + (size_t)base + (size_t)(4 * lane);
    #pragma unroll
    for (int k = 3; k >= 0; --k) {
      lr_in = fmaf(af[k], lr_in, uf[k]);
      sr_in = fmaf(af[k], sr_in, rr[k]);
      // (T+1)-pitched rows: only b32-aligned; NT keeps the write stream
      // from evicting the (L2-resident) read stream.
      __builtin_nontemporal_store(lr_in, &out_lr[ob + k]);
      srv[k] = sr_in;
    }
    // sum_rewards rows are 16B-aligned: single NT b128 store per lane
    v4f s4 = {srv[0], srv[1], srv[2], srv[3]};
    __builtin_nontemporal_store(
        s4, (v4f*)(out_sr + (size_t)row_rt + base) + lane);

    lr_c = __shfl(lr_in, 0, 32);   // lane 0 holds lr/sr at t = base
    sr_c = __shfl(sr_in, 0, 32);
    stg ^= 1;
  }
}

// Fallback for T % 128 != 0 or T > TD_MAX_T (not hit with harness shapes).
__global__ void td_naive_kernel(
    const float* __restrict__ values, const float* __restrict__ rewards,
    const float* __restrict__ dones, const float* __restrict__ raw_gamma,
    const float* __restrict__ raw_lambd, const int* __restrict__ start_idx,
    float* __restrict__ out, int B, int T)
{
  const int row = blockIdx.x * blockDim.x + threadIdx.x;
  if (row >= B) return;
  const float gamma = fmaxf(tanhf(raw_gamma[0]), 0.0f);
  const int s0 = start_idx[0];
  const size_t rv = (size_t)row * (size_t)(T + 1);
  const size_t rt = (size_t)row * (size_t)T;
  float* out_sr = out + (size_t)B * (size_t)(T + 1);
  float lr = values[rv + T], sr = 0.0f;
  out[rv + T] = lr;
  for (int t = T - 1; t >= 0; --t) {
    const float lam = fmaxf(tanhf(raw_lambd[s0 + t]), 0.0f);
    const float cc  = gamma * (1.0f - dones[rt + t]);
    const float a   = cc * lam;
    const float r   = rewards[rt + t];
    sr = fmaf(a, sr, r);
    lr = fmaf(a, lr, fmaf(cc * (1.0f - lam), values[rv + t + 1], r));
    out[rv + t]    = lr;
    out_sr[rt + t] = sr;
  }
}

extern "C" void kernel_launch(void* const* d_in, const int* in_sizes, int n_in,
                              void* d_out, int out_size, void* d_ws, size_t ws_size,
                              hipStream_t stream) {
  const float* values    = (const float*)d_in[0];
  const float* rewards   = (const float*)d_in[1];
  const float* dones     = (const float*)d_in[2];
  const float* raw_gamma = (const float*)d_in[3];
  const float* raw_lambd = (const float*)d_in[4];
  const int*   start_idx = (const int*)d_in[5];
  // end_idx (d_in[6]) is implied by the scan length T (= rewards' time dim).

  const int B = in_sizes[0] - in_sizes[1];   // B*(T+1) - B*T
  const int T = in_sizes[1] / B;

  if ((T & 127) == 0 && T <= TD_MAX_T) {
    const int blocks = (B + 7) / 8;          // 8 wave32 rows per 256-thread block
    td_scan_kernel<<<blocks, 256, 0, stream>>>(
        values, rewards, dones, raw_gamma, raw_lambd, start_idx,
        (float*)d_out, B, T);
  } else {
    const int threads = 256;
    const int blocks  = (B + threads - 1) / threads;
    td_naive_kernel<<<blocks, threads, 0, stream>>>(
        values, rewards, dones, raw_gamma, raw_lambd, start_idx,
        (float*)d_out, B, T);
  }
}